// Gaussian_82325933130210
// MI455X (gfx1250) — compile-verified
//
#include <hip/hip_runtime.h>

typedef __attribute__((ext_vector_type(2))) float v2f;
typedef __attribute__((ext_vector_type(8))) float v8f;

namespace {
constexpr int HW   = 256;
constexpr int IMG  = HW * HW;
constexpr int TILE = 16;
constexpr int HALO = 4;
constexpr int INH  = TILE + 2 * HALO;   // 24-wide halo tile
constexpr int PITCH = 26;               // padded (even -> 8B-aligned float2, no bank clash)
constexpr int XT_F = INH * PITCH;       // 624 floats: input tile, column-major xt[c][r]
constexpr int VT_F = TILE * PITCH;      // 416 floats: vertical result, row-major vt[m][c]
constexpr int WAVE_F = XT_F + VT_F;     // 1040 (even -> preserves 8B alignment per wave)
constexpr int NW   = 8;                 // waves per block
constexpr int GT   = 16;                // tap table (9 taps + zero pad)
constexpr int TILES_PER_IMG = (HW / TILE) * (HW / TILE);  // 256
constexpr int NTILES = 16 * 64 * TILES_PER_IMG;           // 262144
}

__global__ __launch_bounds__(256) void
gauss9_sep_wmma(const float* __restrict__ x, const float* __restrict__ w9,
                float* __restrict__ out) {
  __shared__ float lds[GT + NW * WAVE_F];   // ~33.3 KB of the 320 KB WGP LDS
  const int tid  = threadIdx.x;
  const int lane = tid & 31;
  const int wave = tid >> 5;

  // Recover normalized 1D taps from the provided 2D kernel's first row:
  // w2d[0][d] = a[d]/S^2 with a[0]=1  =>  t[d] = w2d[0][d] / sqrt(w2d[0][0]) = a[d]/S.
  // Entries 9..15 stay 0 (used as the out-of-band value).
  if (tid < GT) {
    float v = 0.f;
    if (tid < 9) v = w9[tid] / sqrtf(w9[0]);
    lds[tid] = v;
  }
  __syncthreads();

  const int p  = lane & 15;          // M for A-operands / N for B-operands
  const int kh = (lane >> 4) << 1;   // K half-select: lanes16-31 carry K+2,K+3

  // Banded weight operand, shared by both passes:
  //   vertical A[m,k]  = t[k-m]  (lane m = p)
  //   horizontal B[k,n] = t[k-n] (lane n = p)
  // Per-lane value in both cases: t[4*kc + kh - p] (0 outside the 9-tap band).
  v2f wband[6];
#pragma unroll
  for (int kc = 0; kc < 6; ++kc) {
    int d0 = kc * 4 + kh - p;
    unsigned c0 = ((unsigned)d0       > 8u) ? 9u : (unsigned)d0;
    unsigned c1 = ((unsigned)(d0 + 1) > 8u) ? 9u : (unsigned)(d0 + 1);
    wband[kc].x = lds[c0];
    wband[kc].y = lds[c1];
  }

  float* xt = lds + GT + wave * WAVE_F;  // xt[c*PITCH + r], c=0..23, r=0..23
  float* vt = xt + XT_F;                 // vt[m*PITCH + c], m=0..15, c=0..23

  const int tileIdx = blockIdx.x * NW + wave;      // grid sized exactly: no bounds check
  const int img = tileIdx >> 8;                    // 256 tiles per image
  const int t   = tileIdx & 255;
  const int i0  = (t >> 4) * TILE;
  const int j0  = (t & 15) * TILE;
  const float* src = x   + (size_t)img * IMG;
  float*       dst = out + (size_t)img * IMG;

  // ---- Phase 1: clamped (replication-pad) 24x24 halo tile -> LDS, column-major
  //      576 elements = 18 iters * 32 lanes; consecutive lanes hit consecutive cols.
#pragma unroll
  for (int it = 0; it < 18; ++it) {
    int f = it * 32 + lane;          // 0..575
    int r = f / INH;
    int c = f - r * INH;
    int gr = min(max(i0 - HALO + r, 0), HW - 1);
    int gc = min(max(j0 - HALO + c, 0), HW - 1);
    xt[c * PITCH + r] = src[gr * HW + gc];
  }
  __syncthreads();

  // ---- Phase 2: vertical pass  V[m][c] = sum_k t[k-?]*X[i0-4+k][j0-4+c], c=0..23
  //      as two 16-col WMMA blocks (c=0..15 and c=8..23; overlap rewrites same values).
#pragma unroll
  for (int cb = 0; cb < 2; ++cb) {
    const int cbase = cb * 8;
    v8f acc = {};
#pragma unroll
    for (int kc = 0; kc < 6; ++kc) {
      // B[k, n] = X[row=k][col=cbase+n]; xt column-major -> aligned 8B pair
      v2f b = *(const v2f*)(xt + (cbase + p) * PITCH + kc * 4 + kh);
      acc = __builtin_amdgcn_wmma_f32_16x16x4_f32(
          /*neg_a=*/false, wband[kc], /*neg_b=*/false, b,
          /*c_mod=*/(short)0, acc, /*reuse_a=*/false, /*reuse_b=*/false);
    }
    // D layout: VGPR v <-> row m = v + 8*(lane>>4); col = lane&15 (+cbase)
#pragma unroll
    for (int v = 0; v < 8; ++v)
      vt[(v + kh * 4) * PITCH + cbase + p] = acc[v];
  }
  __syncthreads();

  // ---- Phase 3: horizontal pass  O = V(16x24) * Wband^T(24x16)
  v8f o = {};
#pragma unroll
  for (int kc = 0; kc < 6; ++kc) {
    // A[m, k] = V[m][k]; vt row-major -> aligned 8B pair
    v2f a = *(const v2f*)(vt + p * PITCH + kc * 4 + kh);
    o = __builtin_amdgcn_wmma_f32_16x16x4_f32(
        false, a, false, wband[kc], (short)0, o, false, false);
  }

  // ---- Phase 4: stream the 16x16 tile out (non-temporal: keep L2 for input halos)
#pragma unroll
  for (int v = 0; v < 8; ++v)
    __builtin_nontemporal_store(o[v], dst + (i0 + v + kh * 4) * HW + (j0 + p));
}

extern "C" void kernel_launch(void* const* d_in, const int* in_sizes, int n_in,
                              void* d_out, int out_size, void* d_ws, size_t ws_size,
                              hipStream_t stream) {
  (void)in_sizes; (void)n_in; (void)d_ws; (void)ws_size; (void)out_size;
  const float* x  = (const float*)d_in[0];   // [16,64,256,256] fp32
  const float* w9 = (const float*)d_in[1];   // [64,1,9,9] fp32; first row suffices
  float* out = (float*)d_out;                // [16,64,256,256] fp32
  dim3 grid(NTILES / NW);                    // 32768 blocks x 256 threads (8 waves)
  gauss9_sep_wmma<<<grid, 256, 0, stream>>>(x, w9, out);
}